// L1RegLoss_27350351741519
// MI455X (gfx1250) — compile-verified
//
#include <hip/hip_runtime.h>
#include <math.h>

typedef __attribute__((ext_vector_type(2))) float v2f;
typedef __attribute__((ext_vector_type(8))) float v8f;

#define N_MAX      20
#define NPAIR      190           // N_MAX*(N_MAX-1)/2
#define REG_WEIGHT 0.01f
#define K1_BLOCKS  2048
#define K1_THREADS 256

// ---------------------------------------------------------------------------
// Wave32 sum reduction through V_WMMA_F32_16X16X4_F32 with B = ones.
// A layout (16x4 f32): lanes 0-15 hold {K=0,K=1}, lanes 16-31 hold {K=2,K=3}.
// We put s in a[0], 0 in a[1]  ->  A[m][0]=s_m, A[m][2]=s_{m+16}.
// With B all-ones: D[m][n] = s_m + s_{m+16} for every n.
// C/D layout: VGPR j = row j (lanes 0-15) / row j+8 (lanes 16-31), so summing
// the 8 D components gives half-sum per lane; shfl_xor(16) completes it.
// EXEC must be all ones here (no divergence at call sites).
// ---------------------------------------------------------------------------
__device__ __forceinline__ float wave_sum_wmma(float s) {
    v2f a; a[0] = s;    a[1] = 0.0f;
    v2f b; b[0] = 1.0f; b[1] = 1.0f;
    v8f c = {0.f, 0.f, 0.f, 0.f, 0.f, 0.f, 0.f, 0.f};
    v8f d = __builtin_amdgcn_wmma_f32_16x16x4_f32(
        /*neg_a=*/false, a, /*neg_b=*/false, b,
        /*c_mod=*/(short)0, c, /*reuse_a=*/false, /*reuse_b=*/false);
    float r = ((d[0] + d[1]) + (d[2] + d[3])) + ((d[4] + d[5]) + (d[6] + d[7]));
    r += __shfl_xor(r, 16, 32);   // combine rows 0-7 half with rows 8-15 half
    return r;                     // full 32-lane sum in every lane
}

// ---------------------------------------------------------------------------
// Kernel 1: streaming |t-p| partial sums. 268 MB of reads -> HBM-bound
// (~11.5 us floor at 23.3 TB/s). float4 (b128) loads, 16 vec4/thread,
// WMMA wave reduce, fixed-order block combine. No atomics => deterministic.
// ---------------------------------------------------------------------------
__global__ __launch_bounds__(K1_THREADS)
void l1_partial_kernel(const float4* __restrict__ t,
                       const float4* __restrict__ p,
                       float* __restrict__ partial,
                       unsigned int n4) {
    unsigned int tid    = blockIdx.x * blockDim.x + threadIdx.x;
    unsigned int stride = gridDim.x * blockDim.x;
    float acc = 0.0f;
    for (unsigned int i = tid; i < n4; i += stride) {
        __builtin_prefetch(&t[i + stride], 0, 0);   // global_prefetch_b8
        __builtin_prefetch(&p[i + stride], 0, 0);
        float4 a = t[i];
        float4 b = p[i];
        acc += fabsf(a.x - b.x) + fabsf(a.y - b.y)
             + fabsf(a.z - b.z) + fabsf(a.w - b.w);
    }
    float wsum = wave_sum_wmma(acc);

    __shared__ float ws[K1_THREADS / 32];
    unsigned int lane = threadIdx.x & 31u;
    unsigned int wave = threadIdx.x >> 5;
    if (lane == 0) ws[wave] = wsum;
    __syncthreads();
    if (threadIdx.x == 0) {
        float s = 0.0f;
        #pragma unroll
        for (int w = 0; w < K1_THREADS / 32; ++w) s += ws[w];   // fixed order
        partial[blockIdx.x] = s;
    }
}

// ---------------------------------------------------------------------------
// Kernel 2: single workgroup finalize.
//   (a) deterministic tree-sum of the 2048 block partials -> l1
//   (b) 20 argmax rounds over latent[G] with exclusion list (tie->lower idx)
//   (c) gather top-20 xyz from R_xyz [3,G], 190 pairwise distances,
//       std with ddof=1, write (total, l1, 0.01*std)
// ---------------------------------------------------------------------------
__global__ __launch_bounds__(1024)
void finalize_kernel(const float* __restrict__ partial, int nparts,
                     const float* __restrict__ latent, int G,
                     const float* __restrict__ rxyz,
                     float* __restrict__ out, float inv_n) {
    __shared__ float svals[1024];
    __shared__ int   sidx[1024];
    __shared__ int   excl[N_MAX];
    __shared__ float sx[N_MAX], sy[N_MAX], sz[N_MAX];
    __shared__ float sdist[NPAIR];
    __shared__ float s_l1;
    int tid = threadIdx.x;

    // ---- (a) reduce block partials, fixed-order tree ----
    float s = 0.0f;
    for (int i = tid; i < nparts; i += 1024) s += partial[i];
    svals[tid] = s;
    __syncthreads();
    for (int st = 512; st > 0; st >>= 1) {
        if (tid < st) svals[tid] += svals[tid + st];
        __syncthreads();
    }
    if (tid == 0) s_l1 = svals[0] * inv_n;
    __syncthreads();

    // ---- (b) iterative top-20 argmax ----
    for (int r = 0; r < N_MAX; ++r) {
        float best = -INFINITY;
        int   bidx = 0x7fffffff;
        for (int i = tid; i < G; i += 1024) {
            bool skip = false;
            for (int e = 0; e < r; ++e) if (excl[e] == i) skip = true;
            if (skip) continue;
            float v = latent[i];
            if (v > best || (v == best && i < bidx)) { best = v; bidx = i; }
        }
        svals[tid] = best;
        sidx[tid]  = bidx;
        __syncthreads();
        for (int st = 512; st > 0; st >>= 1) {
            if (tid < st) {
                float v1 = svals[tid],      v2 = svals[tid + st];
                int   i1 = sidx[tid],       i2 = sidx[tid + st];
                if (v2 > v1 || (v2 == v1 && i2 < i1)) {
                    svals[tid] = v2; sidx[tid] = i2;
                }
            }
            __syncthreads();
        }
        if (tid == 0) excl[r] = sidx[0];
        __syncthreads();
    }

    // gather xyz of top-20 (R_xyz is [3, G] row-major)
    if (tid < N_MAX) {
        int idx = excl[tid];
        sx[tid] = rxyz[idx];
        sy[tid] = rxyz[G + idx];
        sz[tid] = rxyz[2 * G + idx];
    }
    __syncthreads();

    // ---- (c) upper-triangle pairwise distances ----
    for (int p = tid; p < NPAIR; p += 1024) {
        int i = 0, rem = p;
        while (rem >= (N_MAX - 1 - i)) { rem -= (N_MAX - 1 - i); ++i; }
        int j = i + 1 + rem;
        float dx = sx[i] - sx[j];
        float dy = sy[i] - sy[j];
        float dz = sz[i] - sz[j];
        sdist[p] = sqrtf(dx * dx + dy * dy + dz * dz);
    }
    __syncthreads();

    if (tid == 0) {
        float m = 0.0f;
        for (int p = 0; p < NPAIR; ++p) m += sdist[p];
        m /= (float)NPAIR;
        float ss = 0.0f;
        for (int p = 0; p < NPAIR; ++p) { float d = sdist[p] - m; ss += d * d; }
        float stdv = sqrtf(ss / (float)(NPAIR - 1));   // ddof=1
        float reg  = REG_WEIGHT * stdv;
        float l1   = s_l1;
        out[0] = l1 + reg;
        out[1] = l1;
        out[2] = reg;
    }
}

// ---------------------------------------------------------------------------
extern "C" void kernel_launch(void* const* d_in, const int* in_sizes, int n_in,
                              void* d_out, int out_size, void* d_ws, size_t ws_size,
                              hipStream_t stream) {
    const float* target = (const float*)d_in[0];   // [32,512,2048] f32
    const float* pred   = (const float*)d_in[1];   // [32,512,2048] f32
    const float* latent = (const float*)d_in[2];   // [64800] f32
    const float* rxyz   = (const float*)d_in[3];   // [3,64800] f32
    float*       out    = (float*)d_out;           // 3 f32: total, l1, reg
    float*       partial = (float*)d_ws;           // 2048 f32 scratch

    unsigned int n  = (unsigned int)in_sizes[0];   // 33554432, divisible by 4
    unsigned int n4 = n / 4u;
    int          G  = in_sizes[2];                 // 64800

    l1_partial_kernel<<<K1_BLOCKS, K1_THREADS, 0, stream>>>(
        (const float4*)target, (const float4*)pred, partial, n4);

    finalize_kernel<<<1, 1024, 0, stream>>>(
        partial, K1_BLOCKS, latent, G, rxyz, out, 1.0f / (float)n);
}